// EC2OD_85469849190488
// MI455X (gfx1250) — compile-verified
//
#include <hip/hip_runtime.h>
#include <hip/hip_bf16.h>

// Shapes fixed by the reference
#define B_  256
#define M_  512
#define D_  512
#define C_  8
// UNITS = [512, 256, 128]

typedef __attribute__((ext_vector_type(2))) float v2f;
typedef __attribute__((ext_vector_type(8))) float v8f;

// ---------------------------------------------------------------------------
// Kernel 1: masked sum pooling.  pooled[b,d] = sum_m ec[b,m,d] * (mask[b,m]==0)
// Bandwidth-bound: streams the full 268 MB ec_output exactly once, coalesced.
// grid = B*2 blocks of 256 threads; each block owns one b and half of D.
// ---------------------------------------------------------------------------
__global__ __launch_bounds__(256)
void pool_kernel(const float* __restrict__ ec, const float* __restrict__ mask,
                 float* __restrict__ pooled) {
    const int b = blockIdx.x >> 1;
    const int d = ((blockIdx.x & 1) << 8) + threadIdx.x;

    __shared__ float mrow[M_];
    for (int m = threadIdx.x; m < M_; m += 256)
        mrow[m] = (mask[(long)b * M_ + m] == 0.0f) ? 1.0f : 0.0f;
    __syncthreads();

    const float* e = ec + ((long)b * M_) * D_ + d;
    float s = 0.0f;
    #pragma unroll 4
    for (int m = 0; m < M_; ++m)
        s += e[(long)m * D_] * mrow[m];
    pooled[(long)b * D_ + d] = s;
}

// ---------------------------------------------------------------------------
// Kernel 2: H = relu(A @ W[c] + b[c]) using fp32 WMMA (v_wmma_f32_16x16x4_f32).
// One wave32 per 16x16 output tile.  A: (256,K) row-major (optionally per-class
// via strideAclass), W: (C,K,N) row-major, H: (C,256,N).
// A fragment (16x4 f32): lane<16 -> {K=k,K=k+1}, lane>=16 -> {K=k+2,K=k+3}.
// C/D fragment: VGPR r holds row (r + 8*(lane>=16)), col = lane&15.
// ---------------------------------------------------------------------------
__global__ __launch_bounds__(128)
void gemm_bias_relu(const float* __restrict__ A, const float* __restrict__ W,
                    const float* __restrict__ bias, float* __restrict__ H,
                    int K, int N, long strideAclass) {
    const int lane   = threadIdx.x & 31;
    const int wv     = threadIdx.x >> 5;
    const int tilesN = N >> 4;
    const int tileId = blockIdx.x * 4 + wv;
    const int tm = tileId / tilesN;
    const int tn = tileId % tilesN;
    const int c  = blockIdx.z;

    const float* Ac = A + (long)c * strideAclass;
    const float* Wc = W + (long)c * K * N;
    const float* bc = bias + (long)c * N;
    float*       Hc = H + (long)c * B_ * N;

    const int l15  = lane & 15;
    const int half = lane >> 4;       // 0 or 1
    const int kOff = half * 2;        // K sub-offset for this half-wave
    const int row  = tm * 16 + l15;   // A-fragment row
    const int col  = tn * 16 + l15;   // B-fragment / output column

    v8f acc = {};
    for (int k = 0; k < K; k += 4) {
        v2f a = *(const v2f*)(Ac + (long)row * K + k + kOff);
        v2f b;
        b[0] = Wc[(long)(k + kOff)     * N + col];
        b[1] = Wc[(long)(k + kOff + 1) * N + col];
        acc = __builtin_amdgcn_wmma_f32_16x16x4_f32(
                  false, a, false, b, (short)0, acc, false, false);
    }

    const int   mb = tm * 16 + half * 8;
    const float bb = bc[col];
    #pragma unroll
    for (int r = 0; r < 8; ++r) {
        float v = acc[r] + bb;
        Hc[(long)(mb + r) * N + col] = (v > 0.0f) ? v : 0.0f;
    }
}

// ---------------------------------------------------------------------------
// Kernel 3: z = h3 / max(||h3||, eps) along last dim (128).  One wave per row.
// ---------------------------------------------------------------------------
__global__ __launch_bounds__(32)
void l2norm_kernel(const float* __restrict__ h3, float* __restrict__ z) {
    const int  row  = blockIdx.x;           // c*B + b, 0..2047
    const int  lane = threadIdx.x;
    const long base = (long)row * 128 + lane * 4;

    const float4 v = *(const float4*)(h3 + base);
    float ss = v.x * v.x + v.y * v.y + v.z * v.z + v.w * v.w;
    #pragma unroll
    for (int m = 16; m >= 1; m >>= 1) ss += __shfl_xor(ss, m);

    const float r = rsqrtf(fmaxf(ss, 1e-12f));
    float4 o; o.x = v.x * r; o.y = v.y * r; o.z = v.z * r; o.w = v.w * r;
    *(float4*)(z + base) = o;
}

// ---------------------------------------------------------------------------
// Kernel 4: contrastive loss partials.  grid=(16 strips, C classes), block=32.
// Each wave computes 16 rows of ip = (z z^T)*(1-eye)/TEMP for its class via
// fp32 WMMA tiles, folding each 16x16 tile immediately into per-row
// sum(exp(ip)), sum(ip*mask1), sum(ip*mask0).  No B x B tensor materialized.
// ---------------------------------------------------------------------------
__global__ __launch_bounds__(32)
void loss_kernel(const float* __restrict__ z, const int* __restrict__ label,
                 float* __restrict__ partials) {
    const int tm = blockIdx.x;      // 16-row strip
    const int c  = blockIdx.y;
    const int lane = threadIdx.x;
    const float* zc = z + (long)c * B_ * 128;

    // num_p for this class: sum of labels over the batch (labels are 0/1)
    int cnt = 0;
    #pragma unroll
    for (int i = 0; i < 8; ++i) cnt += label[(lane + i * 32) * C_ + c];
    #pragma unroll
    for (int m = 16; m >= 1; m >>= 1) cnt += __shfl_xor(cnt, m);
    const float num_p = (float)cnt;
    const float num_n = (float)B_ - num_p;

    const int l15   = lane & 15;
    const int half  = lane >> 4;
    const int kOff  = half * 2;
    const int rowA  = tm * 16 + l15;        // A-fragment row
    const int mbase = tm * 16 + half * 8;   // first row held by this half-wave

    float sumExp[8], sumM1[8], sumM0[8];
    #pragma unroll
    for (int r = 0; r < 8; ++r) { sumExp[r] = 0.f; sumM1[r] = 0.f; sumM0[r] = 0.f; }

    for (int tn = 0; tn < 16; ++tn) {
        const int col = tn * 16 + l15;
        v8f acc = {};
        for (int k = 0; k < 128; k += 4) {
            v2f a = *(const v2f*)(zc + (long)rowA * 128 + k + kOff);
            v2f b = *(const v2f*)(zc + (long)col  * 128 + k + kOff); // B[k][n]=z[n][k]
            acc = __builtin_amdgcn_wmma_f32_16x16x4_f32(
                      false, a, false, b, (short)0, acc, false, false);
        }
        const float m1 = (float)label[col * C_ + c];
        const float m0 = 1.0f - m1;
        #pragma unroll
        for (int r = 0; r < 8; ++r) {
            const int row = mbase + r;
            float s = (col == row) ? 0.0f : acc[r] * 2.0f;  // *(1-eye)/TEMP, TEMP=0.5
            sumExp[r] += expf(s);
            sumM1[r]  += s * m1;
            sumM0[r]  += s * m0;
        }
    }

    // Reduce across the 16 lanes of each half-wave (columns of the strip)
    #pragma unroll
    for (int r = 0; r < 8; ++r) {
        #pragma unroll
        for (int m = 8; m >= 1; m >>= 1) {
            sumExp[r] += __shfl_xor(sumExp[r], m);
            sumM1[r]  += __shfl_xor(sumM1[r], m);
            sumM0[r]  += __shfl_xor(sumM0[r], m);
        }
    }

    // Per-row negated contribution: log(sum exp) - numer/num_vec
    float part = 0.0f;
    #pragma unroll
    for (int r = 0; r < 8; ++r) {
        const int   row   = mbase + r;
        const int   lab   = label[row * C_ + c];
        const float numer = lab ? sumM1[r] : sumM0[r];
        const float nv    = lab ? num_p : num_n;
        part += logf(sumExp[r]) - numer / nv;
    }
    part += __shfl_xor(part, 16);   // combine the two half-waves (disjoint rows)
    if (lane == 0) partials[c * 16 + tm] = part;
}

// ---------------------------------------------------------------------------
// Kernel 5: deterministic final reduction of the 16 strip partials per class.
// ---------------------------------------------------------------------------
__global__ void loss_reduce(const float* __restrict__ partials,
                            float* __restrict__ losses) {
    const int c = threadIdx.x;
    if (c < C_) {
        float s = 0.0f;
        for (int t = 0; t < 16; ++t) s += partials[c * 16 + t];
        losses[c] = s;   // partials already carry the minus sign
    }
}

// ---------------------------------------------------------------------------
extern "C" void kernel_launch(void* const* d_in, const int* in_sizes, int n_in,
                              void* d_out, int out_size, void* d_ws, size_t ws_size,
                              hipStream_t stream) {
    (void)in_sizes; (void)n_in; (void)out_size; (void)ws_size;

    const float* ec   = (const float*)d_in[0];
    const float* mask = (const float*)d_in[1];
    const int*   lab  = (const int*)  d_in[2];
    const float* W1   = (const float*)d_in[3];
    const float* b1   = (const float*)d_in[4];
    const float* W2   = (const float*)d_in[5];
    const float* b2   = (const float*)d_in[6];
    const float* W3   = (const float*)d_in[7];
    const float* b3   = (const float*)d_in[8];

    float* ws       = (float*)d_ws;
    float* pooled   = ws;                              // 256*512            = 131072 f
    float* h1       = pooled + (long)B_ * D_;          // 8*256*512          = 1048576 f
    float* h2       = h1 + (long)C_ * B_ * 512;        // 8*256*256          = 524288 f
    float* h3       = h2 + (long)C_ * B_ * 256;        // 8*256*128          = 262144 f
    float* partials = h3 + (long)C_ * B_ * 128;        // 128 f
    float* z        = (float*)d_out;                   // C*B*128
    float* losses   = z + (long)C_ * B_ * 128;         // C

    // 1) masked sum pooling (HBM-bound, ~268 MB streamed)
    pool_kernel<<<B_ * 2, 256, 0, stream>>>(ec, mask, pooled);

    // 2) per-class 3-layer ReLU MLP via fp32 WMMA
    //    grid.x = (Mtiles * Ntiles) / 4 waves-per-block
    gemm_bias_relu<<<dim3(128, 1, C_), 128, 0, stream>>>(pooled, W1, b1, h1,
                                                         512, 512, 0L);
    gemm_bias_relu<<<dim3( 64, 1, C_), 128, 0, stream>>>(h1, W2, b2, h2,
                                                         512, 256, (long)B_ * 512);
    gemm_bias_relu<<<dim3( 32, 1, C_), 128, 0, stream>>>(h2, W3, b3, h3,
                                                         256, 128, (long)B_ * 256);

    // 3) l2-normalize into d_out
    l2norm_kernel<<<C_ * B_, 32, 0, stream>>>(h3, z);

    // 4) fused z z^T + supervised-contrastive reductions (WMMA fp32)
    loss_kernel<<<dim3(16, C_), 32, 0, stream>>>(z, lab, partials);

    // 5) deterministic final sum
    loss_reduce<<<1, 32, 0, stream>>>(partials, losses);
}